// GMMflow_fast_64450279244472
// MI455X (gfx1250) — compile-verified
//
#include <hip/hip_runtime.h>
#include <hip/hip_bf16.h>

typedef __attribute__((ext_vector_type(2))) float v2f;
typedef __attribute__((ext_vector_type(4))) float v4f;
typedef __attribute__((ext_vector_type(8))) float v8f;

#define BN 2048
#define MM 1024
#define ND 64

// ---------------------------------------------------------------------------
// Zero the accumulator region of workspace (u_num + norm, contiguous).
// ---------------------------------------------------------------------------
__global__ void gmm_zero_kernel(float* p, int n) {
  int i = blockIdx.x * blockDim.x + threadIdx.x;
  if (i < n) p[i] = 0.0f;
}

// ---------------------------------------------------------------------------
// Per-(m,d) precompute: builds GEMM operands in WMMA-fragment-friendly,
// pair-interleaved layouts so the main kernel can use b64/b128 loads.
//   Bp[k2][2m+j]  (k2=k/2, j=k&1): k<64 -> invSigma[m,k]; k>=64 -> -2*Mut*invS
//   PQ[m2][d]     (float4, m2=m/2): {P[m][d], P[m+1][d], Q[m][d], Q[m+1][d]}
//                  with P = K, Q = v - K*Mut
//   cM[m] = sum_d ( Mut^2 * invSigma + log Sigma )
// One block of 64 threads per component m; LDS tree-reduce for cM.
// ---------------------------------------------------------------------------
__global__ void gmm_prep_kernel(const float* __restrict__ Mu0,
                                const float* __restrict__ Mu1,
                                const float* __restrict__ S0,
                                const float* __restrict__ S1,
                                const float* __restrict__ tP,
                                const float* __restrict__ eP,
                                float* __restrict__ Bp,
                                float* __restrict__ cM,
                                float* __restrict__ PQ) {
  const int m = blockIdx.x;
  const int d = threadIdx.x;
  const float t   = tP[0];
  const float eps = eP[0];
  const float eps2 = eps * eps;
  const int idx = m * ND + d;

  float s0  = S0[idx],  s1  = S1[idx];
  float mu0 = Mu0[idx], mu1 = Mu1[idx];

  float Ds  = sqrtf(4.0f * s0 * s1 + eps2 * eps2);
  float Cs  = 0.5f * (Ds - eps2);
  float vv  = mu1 - mu0;
  float mut = (1.0f - t) * mu0 + t * mu1;
  float Sg  = (1.0f - t) * (1.0f - t) * s0 + t * t * s1
            + 2.0f * t * (1.0f - t) * (Cs + 0.5f * eps2);
  float Pt  = t * s1 + (1.0f - t) * Cs;
  float Qt  = (1.0f - t) * s0 + t * Cs;
  float St  = Pt - Qt - eps2 * t;
  float Kc  = St / Sg;
  float invS = 1.0f / Sg;

  // Pair-interleaved GEMM1 operand: row pair (k, k+1) adjacent per m.
  Bp[(d >> 1) * (2 * MM) + 2 * m + (d & 1)]        = invS;
  Bp[(32 + (d >> 1)) * (2 * MM) + 2 * m + (d & 1)] = -2.0f * mut * invS;

  // Quad-packed GEMM2 operand: {P[m], P[m+1], Q[m], Q[m+1]} per (m-pair, d).
  float* pq = PQ + ((size_t)(m >> 1) * ND + d) * 4;
  pq[(m & 1)]     = Kc;
  pq[2 + (m & 1)] = vv - Kc * mut;

  __shared__ float red[ND];
  red[d] = mut * mut * invS + logf(Sg);
  __syncthreads();
  for (int s = ND / 2; s > 0; s >>= 1) {
    if (d < s) red[d] += red[d + s];
    __syncthreads();
  }
  if (d == 0) cM[m] = red[0];
}

// ---------------------------------------------------------------------------
// Fused main kernel: one wave (32 threads) per (16-row b-strip, 128-m chunk).
//   GEMM1 (f32 WMMA 16x16x4, K=128): logits -> W tile (C layout)
//   LDS 16x16 transpose (stride 18)  : C layout -> A-fragment layout
//   GEMM2 (f32 WMMA 16x16x4, K=16/tile): accumulate W@P, W@Q, W@1
//   Partial results combined across chunks via f32 global atomics.
// ---------------------------------------------------------------------------
__global__ void __launch_bounds__(32)
gmm_main_kernel(const float* __restrict__ X,
                const float* __restrict__ Lam,
                const float* __restrict__ cM,
                const float* __restrict__ Bp,
                const float* __restrict__ PQ,
                float* __restrict__ uAcc,
                float* __restrict__ nAcc) {
  const int strip = blockIdx.x;          // [0,128): 16 b-rows each
  const int chunk = blockIdx.y;          // [0,8):   128 m each
  const int b0    = strip * 16;
  const int l     = threadIdx.x;
  const int lmod  = l & 15;
  const int lhalf = l >> 4;

  __shared__ float lds[16 * 18];

  // --- Preload A fragments of [X^2 | X], invariant over the m loop. ---
  // A 16x4 f32 layout: lane<16 holds K = k0,k0+1 ; lane>=16 holds K = k0+2,k0+3.
  v2f xa[32];
  const float* xrow = X + (size_t)(b0 + lmod) * ND;
#pragma unroll
  for (int kk = 0; kk < 16; ++kk) {      // feature cols 0..63 : X^2
    int k0 = kk * 4 + lhalf * 2;         // even -> 8B aligned
    v2f xv = *(const v2f*)(xrow + k0);
    v2f tfrag; tfrag[0] = xv[0] * xv[0]; tfrag[1] = xv[1] * xv[1];
    xa[kk] = tfrag;
  }
#pragma unroll
  for (int kk = 16; kk < 32; ++kk) {     // feature cols 64..127 : X
    int k0 = kk * 4 + lhalf * 2 - 64;
    xa[kk] = *(const v2f*)(xrow + k0);
  }

  // Persistent accumulators for GEMM2 (C/D layout).
  v8f accP[4] = {};
  v8f accQ[4] = {};
  v8f accN = {};

  // Synthetic B fragment: ones in column 0 only -> accN col0 = sum_m W.
  v2f bones;
  bones[0] = (lmod == 0) ? 1.0f : 0.0f;
  bones[1] = bones[0];

  for (int mt = 0; mt < 8; ++mt) {
    const int m0 = chunk * 128 + mt * 16;

    // ---- GEMM1: L[16b x 16m] = [X^2|X] @ B, K = 128, b64 fragment loads ----
    v8f L = {};
#pragma unroll
    for (int kk = 0; kk < 32; ++kk) {
      const int pairIdx = kk * 2 + lhalf;   // (krow = kk*4 + lhalf*2) / 2
      v2f bf = *(const v2f*)(Bp + (size_t)pairIdx * (2 * MM) + 2 * (m0 + lmod));
      L = __builtin_amdgcn_wmma_f32_16x16x4_f32(false, xa[kk], false, bf,
                                                (short)0, L, false, false);
    }

    // ---- Elementwise: logw = clip(-0.5*(L + c[m])), W = exp(logw)*Lam ----
    const float cm  = cM[m0 + lmod];
    const float lam = Lam[m0 + lmod];
    float w[8];
#pragma unroll
    for (int r = 0; r < 8; ++r) {
      float lw = -0.5f * (L[r] + cm);
      lw = fminf(50.0f, fmaxf(-50.0f, lw));
      w[r] = __expf(lw) * lam;
    }

    // ---- Transpose W tile via LDS: C layout -> A-fragment layout ----
    __syncthreads();
#pragma unroll
    for (int r = 0; r < 8; ++r)
      lds[(r + lhalf * 8) * 18 + lmod] = w[r];   // [b_local][m_local]
    __syncthreads();

    // ---- GEMM2: accumulate W @ [P | Q | 1], b128 fused P/Q loads ----
#pragma unroll
    for (int kk2 = 0; kk2 < 4; ++kk2) {
      const int km = kk2 * 4 + lhalf * 2;        // even -> aligned ds b64
      v2f wa = *(const v2f*)(&lds[lmod * 18 + km]);
      const int mpair = (m0 >> 1) + kk2 * 2 + lhalf;  // (m0 + km) / 2
#pragma unroll
      for (int n = 0; n < 4; ++n) {
        const int d = n * 16 + lmod;
        v4f pq = *(const v4f*)(PQ + ((size_t)mpair * ND + d) * 4);
        v2f bp; bp[0] = pq[0]; bp[1] = pq[1];
        accP[n] = __builtin_amdgcn_wmma_f32_16x16x4_f32(false, wa, false, bp,
                                                        (short)0, accP[n],
                                                        false, false);
        v2f bq; bq[0] = pq[2]; bq[1] = pq[3];
        accQ[n] = __builtin_amdgcn_wmma_f32_16x16x4_f32(false, wa, false, bq,
                                                        (short)0, accQ[n],
                                                        false, false);
      }
      accN = __builtin_amdgcn_wmma_f32_16x16x4_f32(false, wa, false, bones,
                                                   (short)0, accN,
                                                   false, false);
    }
  }

  // ---- Combine: u_num += X*accP + accQ ; norm += accN ----
#pragma unroll
  for (int r = 0; r < 8; ++r) {
    const int b = b0 + r + lhalf * 8;
#pragma unroll
    for (int n = 0; n < 4; ++n) {
      const int d = n * 16 + lmod;
      float val = X[(size_t)b * ND + d] * accP[n][r] + accQ[n][r];
      atomicAdd(&uAcc[(size_t)b * ND + d], val);
    }
    if (lmod == 0) atomicAdd(&nAcc[b], accN[r]);
  }
}

// ---------------------------------------------------------------------------
// Final divide: u = u_num / norm
// ---------------------------------------------------------------------------
__global__ void gmm_finalize_kernel(const float* __restrict__ uAcc,
                                    const float* __restrict__ nAcc,
                                    float* __restrict__ out) {
  int i = blockIdx.x * blockDim.x + threadIdx.x;
  if (i < BN * ND) out[i] = uAcc[i] / nAcc[i / ND];
}

// ---------------------------------------------------------------------------
extern "C" void kernel_launch(void* const* d_in, const int* in_sizes, int n_in,
                              void* d_out, int out_size, void* d_ws, size_t ws_size,
                              hipStream_t stream) {
  const float* X   = (const float*)d_in[0];
  const float* Mu0 = (const float*)d_in[1];
  const float* Mu1 = (const float*)d_in[2];
  const float* S0  = (const float*)d_in[3];
  const float* S1  = (const float*)d_in[4];
  const float* Lam = (const float*)d_in[5];
  const float* tP  = (const float*)d_in[6];
  const float* eP  = (const float*)d_in[7];
  float* out = (float*)d_out;

  // Workspace layout (floats):
  float* ws   = (float*)d_ws;
  float* Bp   = ws;                          // 64 * 2048   = 131072
  float* cM   = Bp + 64 * 2 * MM;            // 1024
  float* PQ   = cM + MM;                     // 512 * 64 * 4 = 131072
  float* uAcc = PQ + (MM / 2) * ND * 4;      // 2048 * 64   = 131072
  float* nAcc = uAcc + BN * ND;              // 2048
  (void)ws_size; (void)in_sizes; (void)n_in; (void)out_size;

  const int nzero = BN * ND + BN;            // uAcc + nAcc contiguous
  gmm_zero_kernel<<<(nzero + 255) / 256, 256, 0, stream>>>(uAcc, nzero);

  gmm_prep_kernel<<<MM, ND, 0, stream>>>(Mu0, Mu1, S0, S1, tP, eP,
                                         Bp, cM, PQ);

  gmm_main_kernel<<<dim3(BN / 16, 8), 32, 0, stream>>>(X, Lam, cM, Bp,
                                                       PQ, uAcc, nAcc);

  gmm_finalize_kernel<<<(BN * ND + 255) / 256, 256, 0, stream>>>(uAcc, nAcc, out);
}